// GNN_40329742909897
// MI455X (gfx1250) — compile-verified
//
#include <hip/hip_runtime.h>
#include <hip/hip_bf16.h>

typedef __attribute__((ext_vector_type(2))) float v2f;
typedef __attribute__((ext_vector_type(8))) float v8f;

// ---------------------------------------------------------------------------
// WMMA fp32 GEMM: C[N x F] = A[N x K] * W[K x F], K/F compile-time.
// One wave per 16-row stripe; NT = F/16 column tiles per wave so the A
// fragment is reused NT times and there are NT independent WMMA chains.
// V_WMMA_F32_16X16X4_F32: A 16x4 (2 VGPR), B 4x16 (2 VGPR), C/D 16x16 (8 VGPR).
// ---------------------------------------------------------------------------
template<int K, int F>
__global__ __launch_bounds__(32) void gcn_gemm_wmma(
    const float* __restrict__ A, const float* __restrict__ W,
    float* __restrict__ C)
{
    constexpr int NT = F / 16;
    const int row0 = blockIdx.x << 4;
    const int lane = threadIdx.x;          // 0..31 (wave32)
    const int m    = lane & 15;            // A row / B,C col within tile
    const int kk   = (lane >> 4) << 1;     // 0 or 2 (ISA 32-bit A/B fragment layout)

    const float* __restrict__ ap = A + (size_t)(row0 + m) * K + kk;
    const float* __restrict__ wp = W + (size_t)kk * F + m;

    v8f acc[NT];
    #pragma unroll
    for (int t = 0; t < NT; ++t) acc[t] = (v8f){};

    #pragma unroll
    for (int k0 = 0; k0 < K; k0 += 4) {
        const v2f a = *(const v2f*)(ap + k0);        // K=kk..kk+1 pair: one b64 load
        #pragma unroll
        for (int t = 0; t < NT; ++t) {
            v2f b;
            b.x = wp[k0 * F + t * 16];               // immediate offsets (K,F constant)
            b.y = wp[(k0 + 1) * F + t * 16];
            acc[t] = __builtin_amdgcn_wmma_f32_16x16x4_f32(
                /*neg_a=*/false, a, /*neg_b=*/false, b,
                /*c_mod=*/(short)0, acc[t], /*reuse_a=*/false, /*reuse_b=*/false);
        }
    }

    // C/D layout: VGPR v -> M = v + 8*(lane>=16), N = lane&15
    const int mBase = (lane >> 4) << 3;
    float* __restrict__ cp = C + (size_t)(row0 + mBase) * F + m;
    #pragma unroll
    for (int v = 0; v < 8; ++v)
        #pragma unroll
        for (int t = 0; t < NT; ++t)
            cp[(size_t)v * F + t * 16] = acc[t][v];
}

// Scalar tail for N % 16 rows (zero blocks when N is a multiple of 16).
__global__ void gcn_gemm_tail(const float* __restrict__ A, const float* __restrict__ W,
                              float* __restrict__ C, int rowStart, int N, int K, int F)
{
    int idx = blockIdx.x * blockDim.x + threadIdx.x;
    int total = (N - rowStart) * F;
    if (idx >= total) return;
    int r = rowStart + idx / F;
    int f = idx % F;
    float s = 0.0f;
    for (int k = 0; k < K; ++k) s += A[(size_t)r * K + k] * W[(size_t)k * F + f];
    C[(size_t)r * F + f] = s;
}

// ---------------------------------------------------------------------------
// Degree / normalization precompute (invariant across all 5 layers)
// ---------------------------------------------------------------------------
__global__ void gcn_deg_init(float* __restrict__ deg, int N) {
    int i = blockIdx.x * blockDim.x + threadIdx.x;
    if (i < N) deg[i] = 1.0f;              // self-loop weight
}

__global__ void gcn_deg_scatter(const int* __restrict__ dst,
                                const float* __restrict__ w,
                                float* __restrict__ deg, int E) {
    int e = blockIdx.x * blockDim.x + threadIdx.x;
    if (e < E) unsafeAtomicAdd(&deg[dst[e]], w[e]);
}

__global__ void gcn_dinv(float* __restrict__ deg, int N) {
    int i = blockIdx.x * blockDim.x + threadIdx.x;
    if (i < N) {
        float d = deg[i];
        deg[i] = (d > 0.0f) ? __frsqrt_rn(d) : 0.0f;   // becomes dinv in place
    }
}

__global__ void gcn_norm(const int* __restrict__ src, const int* __restrict__ dst,
                         const float* __restrict__ w, const float* __restrict__ dinv,
                         float* __restrict__ norm, int E) {
    int e = blockIdx.x * blockDim.x + threadIdx.x;
    if (e < E) norm[e] = dinv[src[e]] * w[e] * dinv[dst[e]];
}

// ---------------------------------------------------------------------------
// Per-layer aggregation
// ---------------------------------------------------------------------------
// agg[i,:] = dinv[i]^2 * xw[i,:]   (self-loop contribution; also inits agg)
__global__ void gcn_agg_init(const float* __restrict__ xw, const float* __restrict__ dinv,
                             float* __restrict__ agg, int total, int l2f) {
    int g = blockIdx.x * blockDim.x + threadIdx.x;
    if (g < total) {
        int i = g >> l2f;
        float di = dinv[i];
        agg[g] = di * di * xw[g];
    }
}

// One thread per (edge, 4-feature chunk): gather float4 from xw[src], atomic-add to agg[dst]
__global__ void gcn_edge_scatter(const int* __restrict__ src, const int* __restrict__ dst,
                                 const float* __restrict__ norm,
                                 const float* __restrict__ xw, float* __restrict__ agg,
                                 int E, int Fout, int l2c) {
    long long g = (long long)blockIdx.x * blockDim.x + threadIdx.x;
    int chunks = Fout >> 2;
    long long total = (long long)E * chunks;
    if (g >= total) return;
    int e = (int)(g >> l2c);
    int c = (int)(g & (chunks - 1));
    int s = src[e], d = dst[e];
    float nv = norm[e];
    const float4 v = *(const float4*)(xw + (size_t)s * Fout + (c << 2));
    float* p = agg + (size_t)d * Fout + (c << 2);
    unsafeAtomicAdd(p + 0, nv * v.x);
    unsafeAtomicAdd(p + 1, nv * v.y);
    unsafeAtomicAdd(p + 2, nv * v.z);
    unsafeAtomicAdd(p + 3, nv * v.w);
}

// h = (relu?) (agg + b), in place
__global__ void gcn_epilogue(float* __restrict__ agg, const float* __restrict__ b,
                             int total, int fmask, int doRelu) {
    int g = blockIdx.x * blockDim.x + threadIdx.x;
    if (g < total) {
        float v = agg[g] + b[g & fmask];
        agg[g] = doRelu ? fmaxf(v, 0.0f) : v;
    }
}

// ---------------------------------------------------------------------------
// Global add pool + final relu
// ---------------------------------------------------------------------------
__global__ void gcn_zero(float* __restrict__ p, int n) {
    int g = blockIdx.x * blockDim.x + threadIdx.x;
    if (g < n) p[g] = 0.0f;
}

__global__ void gcn_pool(const float* __restrict__ h, const int* __restrict__ batch,
                         float* __restrict__ out, int total) {
    int g = blockIdx.x * blockDim.x + threadIdx.x;
    if (g < total) {
        int i = g >> 7;                    // F = 128
        int f = g & 127;
        unsafeAtomicAdd(&out[((size_t)batch[i] << 7) + f], h[g]);
    }
}

__global__ void gcn_relu(float* __restrict__ p, int n) {
    int g = blockIdx.x * blockDim.x + threadIdx.x;
    if (g < n) p[g] = fmaxf(p[g], 0.0f);
}

// ---------------------------------------------------------------------------
static inline int ceilDiv(long long a, int b) { return (int)((a + b - 1) / b); }

extern "C" void kernel_launch(void* const* d_in, const int* in_sizes, int n_in,
                              void* d_out, int out_size, void* d_ws, size_t ws_size,
                              hipStream_t stream) {
    const float* x     = (const float*)d_in[0];
    const int*   ei    = (const int*)  d_in[1];
    const float* ew    = (const float*)d_in[2];
    const int*   batch = (const int*)  d_in[3];
    const float* Ws[5] = { (const float*)d_in[4],  (const float*)d_in[6],
                           (const float*)d_in[8],  (const float*)d_in[10],
                           (const float*)d_in[12] };
    const float* bs[5] = { (const float*)d_in[5],  (const float*)d_in[7],
                           (const float*)d_in[9],  (const float*)d_in[11],
                           (const float*)d_in[13] };

    const int N = in_sizes[0] / 64;
    const int E = in_sizes[1] / 2;
    const int* src = ei;
    const int* dst = ei + E;

    // Workspace (floats): bufA (xw, N*128) | bufB (agg/h, N*128) | dinv (N) | norm (E)
    float* bufA = (float*)d_ws;
    float* bufB = bufA + (size_t)N * 128;
    float* dinv = bufB + (size_t)N * 128;
    float* norm = dinv + N;

    const int B = 256;
    const int rowTiles = N >> 4;
    const int remRows  = N - (rowTiles << 4);

    // ---- degree / dinv / norm (shared by all layers) ----
    gcn_deg_init   <<<ceilDiv(N, B), B, 0, stream>>>(dinv, N);
    gcn_deg_scatter<<<ceilDiv(E, B), B, 0, stream>>>(dst, ew, dinv, E);
    gcn_dinv       <<<ceilDiv(N, B), B, 0, stream>>>(dinv, N);
    gcn_norm       <<<ceilDiv(E, B), B, 0, stream>>>(src, dst, ew, dinv, norm, E);

    // ---- layers ----
    const int fins [5] = { 64, 16, 32, 64,  64 };
    const int fouts[5] = { 16, 32, 64, 64, 128 };

    for (int l = 0; l < 5; ++l) {
        const int K = fins[l], F = fouts[l];
        const float* hIn = (l == 0) ? x : bufB;
        int l2f = (F == 16) ? 4 : (F == 32) ? 5 : (F == 64) ? 6 : 7;
        int l2c = l2f - 2;                         // chunks = F/4

        if (rowTiles > 0) {
            switch (l) {
                case 0: gcn_gemm_wmma<64, 16><<<rowTiles, 32, 0, stream>>>(hIn, Ws[l], bufA); break;
                case 1: gcn_gemm_wmma<16, 32><<<rowTiles, 32, 0, stream>>>(hIn, Ws[l], bufA); break;
                case 2: gcn_gemm_wmma<32, 64><<<rowTiles, 32, 0, stream>>>(hIn, Ws[l], bufA); break;
                case 3: gcn_gemm_wmma<64, 64><<<rowTiles, 32, 0, stream>>>(hIn, Ws[l], bufA); break;
                case 4: gcn_gemm_wmma<64,128><<<rowTiles, 32, 0, stream>>>(hIn, Ws[l], bufA); break;
            }
        }
        if (remRows > 0) {
            gcn_gemm_tail<<<ceilDiv((long long)remRows * F, B), B, 0, stream>>>(
                hIn, Ws[l], bufA, rowTiles << 4, N, K, F);
        }

        long long totalNF = (long long)N * F;
        gcn_agg_init<<<ceilDiv(totalNF, B), B, 0, stream>>>(bufA, dinv, bufB, (int)totalNF, l2f);

        long long totalEC = (long long)E * (F >> 2);
        gcn_edge_scatter<<<ceilDiv(totalEC, B), B, 0, stream>>>(src, dst, norm, bufA, bufB,
                                                                E, F, l2c);

        gcn_epilogue<<<ceilDiv(totalNF, B), B, 0, stream>>>(bufB, bs[l], (int)totalNF,
                                                            F - 1, (l < 4) ? 1 : 0);
    }

    // ---- global add pool over batch, then relu ----
    float* out = (float*)d_out;
    gcn_zero<<<ceilDiv(out_size, B), B, 0, stream>>>(out, out_size);
    long long totalN128 = (long long)N * 128;
    gcn_pool<<<ceilDiv(totalN128, B), B, 0, stream>>>(bufB, batch, out, (int)totalN128);
    gcn_relu<<<ceilDiv(out_size, B), B, 0, stream>>>(out, out_size);
}